// AttentionGRUCell_1580547966391
// MI455X (gfx1250) — compile-verified
//
#include <hip/hip_runtime.h>
#include <hip/hip_bf16.h>
#include <math.h>

// Problem constants (from reference)
#define BB   256
#define TT   256
#define DD   512   // D_IN
#define HH   512   // hidden
#define NE   97
#define G3   1536  // 3*H
#define LDA  516   // 512 + 4 pad: rows stay 16B aligned, kills DS bank conflicts

typedef __attribute__((ext_vector_type(2))) float v2f;
typedef __attribute__((ext_vector_type(8))) float v8f;

// -------------------------------------------------------------------------
// prev_proj[b,h] = prev_hidden[b,:] . h2h_w[h,:] + h2h_b[h]
__global__ __launch_bounds__(256) void prev_proj_kernel(
    const float* __restrict__ prev_h, const float* __restrict__ h2h_w,
    const float* __restrict__ h2h_b, float* __restrict__ pp) {
  __shared__ float sh[HH];
  const int b = blockIdx.x;
  const int h = blockIdx.y * 256 + threadIdx.x;
  for (int k = threadIdx.x; k < HH; k += 256) sh[k] = prev_h[b * HH + k];
  __syncthreads();
  const float* w = h2h_w + (size_t)h * HH;
  float s = h2h_b[h];
#pragma unroll 8
  for (int k = 0; k < HH; ++k) s += sh[k] * w[k];
  pp[b * HH + h] = s;
}

// -------------------------------------------------------------------------
// Fused WMMA kernel: e[b,t] = sum_h score_w[h]*tanh( (batch_H @ i2h_w^T)[b,t,h]
//                                                    + prev_proj[b,h] )
// One block per (b, 16-row t-tile); 8 waves, each wave owns 4 h-tiles (N dim).
// f32 WMMA 16x16x4: A lanes 0-15 hold M, K={0,1}; lanes 16-31 K={2,3}.
// A tile staged into LDS via CDNA5 async global->LDS copies (ASYNCcnt path).
__global__ __launch_bounds__(256) void attn_score_wmma(
    const float* __restrict__ batch_H, const float* __restrict__ i2h_w,
    const float* __restrict__ prev_proj, const float* __restrict__ score_w,
    float* __restrict__ e_out) {
  __shared__ __align__(16) float lds_a[16 * LDA];   // 33 KB A tile
  __shared__ float e_buf[16];

  const int tid  = threadIdx.x;
  const int wave = tid >> 5;
  const int lane = tid & 31;
  const int b    = blockIdx.x;
  const int t0   = blockIdx.y * 16;

  if (tid < 16) e_buf[tid] = 0.f;

  // Stage A tile batch_H[b, t0:t0+16, 0:512] into LDS with the async copy
  // engine: global_load_async_to_lds_b128 takes a per-lane LDS byte address in
  // VDST, so the 516-float padded row layout costs nothing extra. No VGPR
  // round-trip; completion tracked by ASYNCcnt.
  const float4* src = (const float4*)(batch_H + ((size_t)b * TT + t0) * DD);
  const unsigned lds_base = (unsigned)(uintptr_t)&lds_a[0];
#pragma unroll
  for (int j = 0; j < 8; ++j) {               // 8 * 256 lanes * 16B = 32 KB
    const int idx = tid + j * 256;            // float4 index, 0..2047
    const int row = idx >> 7;                 // /128 float4 per source row
    const int c4  = idx & 127;
    const float4* g = src + row * (DD / 4) + c4;
    const unsigned l = lds_base + (unsigned)((row * LDA + c4 * 4) * 4);
    asm volatile("global_load_async_to_lds_b128 %0, %1, off"
                 :: "v"(l), "v"(g)
                 : "memory");
  }
  asm volatile("s_wait_asynccnt 0" ::: "memory");
  __syncthreads();

  const int tl    = lane & 15;          // row (M) / col (N) within tile
  const int khalf = (lane >> 4) * 2;    // K sub-offset: 0 or 2
  float epart[8] = {0.f, 0.f, 0.f, 0.f, 0.f, 0.f, 0.f, 0.f};

  for (int i = 0; i < 4; ++i) {
    const int h0 = (wave * 4 + i) * 16;
    const int h  = h0 + tl;                              // N index for this lane
    const float* brow = i2h_w + (size_t)h * DD;          // B[k][n] = i2h_w[h][k]
    v8f acc = {};
#pragma unroll 8
    for (int k = 0; k < DD; k += 4) {
      v2f a  = *(const v2f*)&lds_a[tl * LDA + k + khalf];  // ds_load_2addr_b64
      v2f bb = *(const v2f*)&brow[k + khalf];              // global b64, L2-hot
      acc = __builtin_amdgcn_wmma_f32_16x16x4_f32(
          /*neg_a=*/false, a, /*neg_b=*/false, bb,
          /*c_mod=*/(short)0, acc, /*reuse_a=*/false, /*reuse_b=*/false);
    }
    const float pp = prev_proj[b * HH + h];
    const float sw = score_w[h];
#pragma unroll
    for (int r = 0; r < 8; ++r) {
      // D layout: VGPR r, lane L -> (M = r + 8*(L/16), N = L&15)
      float v = tanhf(acc[r] + pp) * sw;
      v += __shfl_xor(v, 8);
      v += __shfl_xor(v, 4);
      v += __shfl_xor(v, 2);
      v += __shfl_xor(v, 1);           // sum over 16-lane N group
      epart[r] += v;
    }
  }
  if ((lane & 15) == 0) {
    const int toff = (lane >> 4) * 8;  // lane 0 -> t rows 0..7, lane 16 -> 8..15
#pragma unroll
    for (int r = 0; r < 8; ++r) atomicAdd(&e_buf[toff + r], epart[r]);  // ds_add_f32
  }
  __syncthreads();
  if (tid < 16) e_out[b * TT + t0 + tid] = e_buf[tid];
}

// -------------------------------------------------------------------------
// Softmax over T per batch row; writes alpha straight into d_out's alpha slab.
__global__ __launch_bounds__(256) void softmax_T(
    const float* __restrict__ e, float* __restrict__ alpha) {
  __shared__ float red[TT];
  const int b = blockIdx.x, t = threadIdx.x;
  const float x = e[b * TT + t];
  red[t] = x;
  __syncthreads();
  for (int s = 128; s > 0; s >>= 1) {
    if (t < s) red[t] = fmaxf(red[t], red[t + s]);
    __syncthreads();
  }
  const float mx = red[0];
  __syncthreads();
  const float ex = expf(x - mx);
  red[t] = ex;
  __syncthreads();
  for (int s = 128; s > 0; s >>= 1) {
    if (t < s) red[t] += red[t + s];
    __syncthreads();
  }
  alpha[b * TT + t] = ex / red[0];
}

// -------------------------------------------------------------------------
// context[b,d] = sum_t alpha[b,t] * batch_H[b,t,d]  (2nd pass; L2-resident)
__global__ __launch_bounds__(512) void context_kernel(
    const float* __restrict__ alpha, const float* __restrict__ batch_H,
    float* __restrict__ ctx) {
  __shared__ float sal[TT];
  const int b = blockIdx.x, d = threadIdx.x;
  if (d < TT) sal[d] = alpha[b * TT + d];
  __syncthreads();
  const float* bp = batch_H + (size_t)b * TT * DD + d;
  float acc = 0.f;
#pragma unroll 4
  for (int t = 0; t < TT; ++t) {
    if (t + 16 < TT) __builtin_prefetch(bp + (size_t)(t + 16) * DD, 0, 1);
    acc += sal[t] * bp[(size_t)t * DD];
  }
  ctx[b * DD + d] = acc;
}

// -------------------------------------------------------------------------
// gi[b,j] = [ctx|onehot] . w_ih[j,:] + b_ih[j];  gh[b,j] = prev_h . w_hh[j,:] + b_hh[j]
__global__ __launch_bounds__(256) void gates_gemv(
    const float* __restrict__ ctx, const float* __restrict__ oneh,
    const float* __restrict__ prev_h,
    const float* __restrict__ w_ih, const float* __restrict__ b_ih,
    const float* __restrict__ w_hh, const float* __restrict__ b_hh,
    float* __restrict__ gi, float* __restrict__ gh) {
  __shared__ float sc[DD];
  __shared__ float sp[HH];
  __shared__ float so[NE];
  const int b = blockIdx.x;
  const int j = blockIdx.y * 256 + threadIdx.x;  // < 1536
  for (int k = threadIdx.x; k < DD; k += 256) sc[k] = ctx[b * DD + k];
  for (int k = threadIdx.x; k < HH; k += 256) sp[k] = prev_h[b * HH + k];
  if (threadIdx.x < NE) so[threadIdx.x] = oneh[b * NE + threadIdx.x];
  __syncthreads();
  const float* wi = w_ih + (size_t)j * (DD + NE);
  const float* wh = w_hh + (size_t)j * HH;
  float s = b_ih[j], s2 = b_hh[j];
#pragma unroll 4
  for (int k = 0; k < DD; ++k) { s += sc[k] * wi[k]; s2 += sp[k] * wh[k]; }
  for (int k = 0; k < NE; ++k) s += so[k] * wi[DD + k];
  gi[b * G3 + j] = s;
  gh[b * G3 + j] = s2;
}

// -------------------------------------------------------------------------
// GRU gate math -> cur_hidden into d_out
__global__ __launch_bounds__(256) void gru_gate(
    const float* __restrict__ gi, const float* __restrict__ gh,
    const float* __restrict__ prev_h, float* __restrict__ outh) {
  const int b = blockIdx.x;
  const int h = blockIdx.y * 256 + threadIdx.x;
  const float i_r = gi[b * G3 + h];
  const float i_z = gi[b * G3 + HH + h];
  const float i_n = gi[b * G3 + 2 * HH + h];
  const float h_r = gh[b * G3 + h];
  const float h_z = gh[b * G3 + HH + h];
  const float h_n = gh[b * G3 + 2 * HH + h];
  const float r = 1.f / (1.f + expf(-(i_r + h_r)));
  const float z = 1.f / (1.f + expf(-(i_z + h_z)));
  const float n = tanhf(i_n + r * h_n);
  outh[b * HH + h] = (1.f - z) * n + z * prev_h[b * HH + h];
}

// -------------------------------------------------------------------------
extern "C" void kernel_launch(void* const* d_in, const int* in_sizes, int n_in,
                              void* d_out, int out_size, void* d_ws, size_t ws_size,
                              hipStream_t stream) {
  (void)in_sizes; (void)n_in; (void)out_size; (void)ws_size;
  const float* prev_hidden = (const float*)d_in[0];
  const float* batch_H     = (const float*)d_in[1];
  const float* char_onehot = (const float*)d_in[2];
  const float* i2h_w       = (const float*)d_in[3];
  const float* h2h_w       = (const float*)d_in[4];
  const float* h2h_b       = (const float*)d_in[5];
  const float* score_w     = (const float*)d_in[6];
  const float* rnn_w_ih    = (const float*)d_in[7];
  const float* rnn_b_ih    = (const float*)d_in[8];
  const float* rnn_w_hh    = (const float*)d_in[9];
  const float* rnn_b_hh    = (const float*)d_in[10];

  float* out_hidden = (float*)d_out;                 // [B*H]
  float* out_alpha  = out_hidden + (size_t)BB * HH;  // [B*T]

  float* ws        = (float*)d_ws;
  float* prev_proj = ws;                             // B*H
  float* e_scores  = prev_proj + (size_t)BB * HH;    // B*T
  float* context   = e_scores + (size_t)BB * TT;     // B*D
  float* gi        = context + (size_t)BB * DD;      // B*3H
  float* gh        = gi + (size_t)BB * G3;           // B*3H

  prev_proj_kernel<<<dim3(BB, HH / 256), 256, 0, stream>>>(
      prev_hidden, h2h_w, h2h_b, prev_proj);

  attn_score_wmma<<<dim3(BB, TT / 16), 256, 0, stream>>>(
      batch_H, i2h_w, prev_proj, score_w, e_scores);

  softmax_T<<<BB, TT, 0, stream>>>(e_scores, out_alpha);

  context_kernel<<<BB, DD, 0, stream>>>(out_alpha, batch_H, context);

  gates_gemv<<<dim3(BB, G3 / 256), 256, 0, stream>>>(
      context, char_onehot, prev_hidden, rnn_w_ih, rnn_b_ih,
      rnn_w_hh, rnn_b_hh, gi, gh);

  gru_gate<<<dim3(BB, HH / 256), 256, 0, stream>>>(gi, gh, prev_hidden, out_hidden);
}